// Dissection_10256381903265
// MI455X (gfx1250) — compile-verified
//
#include <hip/hip_runtime.h>
#include <hip/hip_bf16.h>
#include <math.h>

// Problem constants from the reference
#define B_   128
#define NL_  13
#define S_   128
#define D_   768
#define LSTART 4
#define L_   9          // layers used (13 - 4)
#define T_   127        // tokens used (128 - 1)
#define WIN_ 2

typedef __attribute__((ext_vector_type(2))) float v2f;
typedef __attribute__((ext_vector_type(8))) float v8f;

// ---------------------------------------------------------------------------
// Fully unrolled Cholesky (R^T R = windowed Gram) + align/nov terms.
// M is compile-time (3,4,5) so R[][] stays entirely in VGPRs (static indices).
// Row order matches the reference: window rows excluding k (ascending), then k.
// ---------------------------------------------------------------------------
template <int M>
__device__ __forceinline__ void alpha_terms(const float* __restrict__ G,
                                            int lo, int k,
                                            float& outAlign, float& outNov) {
  int rows[M];
#pragma unroll
  for (int i = 0; i < M - 1; ++i) {
    int v = lo + i;
    rows[i] = v + (v >= k ? 1 : 0);   // skip row k
  }
  rows[M - 1] = k;

  float R[M][M];
#pragma unroll
  for (int j = 0; j < M; ++j) {
#pragma unroll
    for (int i = 0; i <= j; ++i) {
      float s = G[rows[i] * 16 + rows[j]];
#pragma unroll
      for (int p = 0; p < i; ++p) s -= R[p][i] * R[p][j];
      R[i][j] = (i == j) ? sqrtf(fmaxf(s, 0.0f)) : s / R[i][i];
    }
  }

  // r = R[:, M-1]
  float rn2 = 0.0f;                        // ||r[:-1]||^2
#pragma unroll
  for (int i = 0; i < M - 1; ++i) rn2 += R[i][M - 1] * R[i][M - 1];
  const float rlast = R[M - 1][M - 1];
  const float rall  = sqrtf(rn2 + rlast * rlast);

  // column-normalize Rn = R[:-1,:-1]; mean over columns per row
  float meanv[M - 1];
#pragma unroll
  for (int i = 0; i < M - 1; ++i) meanv[i] = 0.0f;
#pragma unroll
  for (int j = 0; j < M - 1; ++j) {
    float c2 = 0.0f;
#pragma unroll
    for (int i = 0; i <= j; ++i) c2 += R[i][j] * R[i][j];
    float cn = sqrtf(c2);
    cn = (cn == 0.0f) ? 1.0f : cn;
    const float icn = 1.0f / cn;
#pragma unroll
    for (int i = 0; i <= j; ++i) meanv[i] += R[i][j] * icn;
  }
  const float invm1 = 1.0f / (float)(M - 1);
  float align = 0.0f;
#pragma unroll
  for (int i = 0; i < M - 1; ++i) align += (meanv[i] * invm1) * R[i][M - 1];
  align /= sqrtf(rn2);

  outAlign = 1.0f / (align * (float)M * 2.0f);
  outNov   = fabsf(rlast) / rall;
}

// ---------------------------------------------------------------------------
// Kernel 1: one wave per token (b,t).
//   G = A * A^T via V_WMMA_F32_16X16X4_F32 (A operand == B operand registers).
//   Padding lanes (rows 9..15) load a CLAMPED valid row unconditionally:
//   G[i][j] for i,j<=8 depends only on A-row i / B-col j, so garbage in the
//   16x16 tile outside the 9x9 region we read is harmless. This keeps the
//   inner loop branch-free so loads pipeline ahead of the WMMAs.
// ---------------------------------------------------------------------------
__global__ __launch_bounds__(256) void gram_alpha_kernel(
    const float* __restrict__ X, float* __restrict__ wbuf,
    float* __restrict__ varsum) {
  __shared__ float Gt[8][256];    // 16x16 Gram tile per wave
  __shared__ float Ared[8][16];
  __shared__ float Nred[8][16];

  const int tid  = threadIdx.x;
  const int wv   = tid >> 5;           // wave id in block (wave32)
  const int lane = tid & 31;

  const int token = blockIdx.x * 8 + wv;   // 2032*8 == 16256 exactly
  const int b = token / T_;
  const int t = token - b * T_;

  // WMMA f32 16x16x4 operand layout (wave32):
  //   lanes 0-15 : row = lane,    K-pair {0,1} of each 4-chunk
  //   lanes16-31 : row = lane-16, K-pair {2,3}
  // Identical layout serves the B operand of A*A^T, so feed `a` twice.
  const int row    = lane & 15;
  const int srow   = (row < L_) ? row : (L_ - 1);  // clamp: always-valid addr
  const int colOff = (lane >> 4) << 1;             // 0 or 2

  const float* base =
      X + ((size_t)(b * NL_ + LSTART + srow) * S_ + t) * D_ + colOff;

  v8f acc = {0.f, 0.f, 0.f, 0.f, 0.f, 0.f, 0.f, 0.f};
#pragma unroll 8
  for (int kb = 0; kb < D_; kb += 4) {
    const v2f a = *(const v2f*)(base + kb);     // unconditional global_load_b64
    // D = A * A^T + C  (emits v_wmma_f32_16x16x4_f32)
    acc = __builtin_amdgcn_wmma_f32_16x16x4_f32(
        false, a, false, a, (short)0, acc, false, false);
  }

  // C/D layout: lanes 0-15 rows 0-7, lanes 16-31 rows 8-15; col = lane&15
  {
    const int mrow = (lane >> 4) * 8;
    const int n = lane & 15;
#pragma unroll
    for (int i = 0; i < 8; ++i) Gt[wv][(mrow + i) * 16 + n] = acc[i];
  }
  __syncthreads();

  const float* G = &Gt[wv][0];

  float alignv = 0.0f, novv = 0.0f;
  if (lane < L_) {
    const int k = lane;
    int lo = k - WIN_; if (lo < 0) lo = 0;
    int hi = k + WIN_ + 1; if (hi > L_) hi = L_;
    const int m = hi - lo;
    if      (m == 3) alpha_terms<3>(G, lo, k, alignv, novv);
    else if (m == 4) alpha_terms<4>(G, lo, k, alignv, novv);
    else             alpha_terms<5>(G, lo, k, alignv, novv);
    Ared[wv][lane] = alignv;
    Nred[wv][lane] = novv;
  }
  __syncthreads();

  float sumA = 0.0f, sumN = 0.0f;
#pragma unroll
  for (int i = 0; i < L_; ++i) { sumA += Ared[wv][i]; sumN += Nred[wv][i]; }

  // variance of adjacent-layer cosine similarities straight from G
  float mean = 0.0f, msq = 0.0f;
#pragma unroll
  for (int l = 0; l < L_ - 1; ++l) {
    const float c = G[l * 16 + l + 1] * rsqrtf(G[l * 17] * G[(l + 1) * 17]);
    mean += c; msq += c * c;
  }
  mean *= (1.0f / (float)(L_ - 1));
  msq  *= (1.0f / (float)(L_ - 1));
  const float var = msq - mean * mean;

  if (lane < L_) {
    // Sum of (aa/sumA + an/sumN) over k is exactly 2 -> normalizer is 0.5
    const float alpha = 0.5f * (alignv / sumA + novv / sumN);
    wbuf[(size_t)token * L_ + lane] = alpha * var;   // var normalized later
  }
  if (lane == 0) atomicAdd(varsum + b, var);
}

// ---------------------------------------------------------------------------
// Kernel 2: out[b,d] = (1/varsum[b]) * sum_t sum_l w[b,t,l] * X[b,4+l,t,d]
// Pure bandwidth pass; t-split across grid.z with f32 atomics into zeroed out.
// ---------------------------------------------------------------------------
__global__ __launch_bounds__(256) void weighted_reduce_kernel(
    const float* __restrict__ X, const float* __restrict__ wbuf,
    const float* __restrict__ varsum, float* __restrict__ out) {
  const int d  = blockIdx.x * 256 + threadIdx.x;   // 3 * 256 = 768
  const int b  = blockIdx.y;
  const int t0 = blockIdx.z * 16;
  int t1 = t0 + 16; if (t1 > T_) t1 = T_;

  const float inv = 1.0f / varsum[b];
  const float* Xb = X + (size_t)(b * NL_ + LSTART) * S_ * D_ + d;
  const float* Wb = wbuf + (size_t)b * T_ * L_;

  float acc = 0.0f;
  for (int t = t0; t < t1; ++t) {
    if (t + 1 < t1)  // gfx1250 global_prefetch_b8 hint for next token row
      __builtin_prefetch(Xb + (size_t)(t + 1) * D_, 0, 1);
    const float* xt = Xb + (size_t)t * D_;
    const float* wt = Wb + t * L_;
#pragma unroll
    for (int l = 0; l < L_; ++l)
      acc = fmaf(wt[l], xt[(size_t)l * S_ * D_], acc);
  }
  atomicAdd(out + (size_t)b * D_ + d, acc * inv);
}

// ---------------------------------------------------------------------------
extern "C" void kernel_launch(void* const* d_in, const int* in_sizes, int n_in,
                              void* d_out, int out_size, void* d_ws,
                              size_t ws_size, hipStream_t stream) {
  const float* X = (const float*)d_in[0];   // (128,13,128,768) f32
  // d_in[1] = masks (all ones in reference; unused by the math)
  float* out = (float*)d_out;               // (128,768) f32

  // workspace layout: [varsum: 128 f32][wbuf: 16256*9 f32]
  float* varsum = (float*)d_ws;
  float* wbuf   = varsum + 128;

  hipMemsetAsync(varsum, 0, 128 * sizeof(float), stream);
  hipMemsetAsync(out, 0, (size_t)out_size * sizeof(float), stream);

  gram_alpha_kernel<<<dim3(2032), dim3(256), 0, stream>>>(X, wbuf, varsum);
  weighted_reduce_kernel<<<dim3(3, B_, 8), dim3(256), 0, stream>>>(
      X, wbuf, varsum, out);
}